// DeciLMMoe_25709674234497
// MI455X (gfx1250) — compile-verified
//
#include <hip/hip_runtime.h>
#include <hip/hip_bf16.h>
#include <math.h>

typedef __attribute__((ext_vector_type(2))) float v2f;
typedef __attribute__((ext_vector_type(8))) float v8f;

#define TB 2
#define SS 2048
#define HH 2048
#define EE 8
#define II 1024
#define ISS 4096
#define TT (TB * SS)   // 4096 tokens

#define BM 64
#define BN 128
#define BK 16
#define LSTR 20        // padded LDS row stride in floats (80B: 16B aligned, conflict-free)

#define WMMA_F32(a, b, c) \
  __builtin_amdgcn_wmma_f32_16x16x4_f32(false, (a), false, (b), (short)0, (c), false, false)

static __device__ __forceinline__ v2f mk2(float2 t) { v2f r; r[0] = t.x; r[1] = t.y; return r; }

// ---------------------------------------------------------------------------
// Router: logits = x @ router_w^T, top-2, sigmoid -> per-token/expert scale,
// plus compacted per-expert token lists for the sparse routed GEMMs.
// One wave per token.
// ---------------------------------------------------------------------------
__global__ __launch_bounds__(256) void router_kernel(
    const float* __restrict__ x, const float* __restrict__ rw,
    float* __restrict__ logits, float* __restrict__ scale,
    int* __restrict__ counts, int* __restrict__ lists)
{
  const int lane = threadIdx.x & 31;
  const int wv   = threadIdx.x >> 5;
  const int t    = blockIdx.x * 8 + wv;
  if (t >= TT) return;

  const float* xp = x + (size_t)t * HH;
  float acc[EE];
#pragma unroll
  for (int e = 0; e < EE; ++e) acc[e] = 0.f;

  for (int h = lane; h < HH; h += 32) {
    const float xv = xp[h];
#pragma unroll
    for (int e = 0; e < EE; ++e) acc[e] += xv * rw[e * HH + h];
  }
#pragma unroll
  for (int e = 0; e < EE; ++e) {
#pragma unroll
    for (int off = 16; off > 0; off >>= 1) acc[e] += __shfl_xor(acc[e], off, 32);
  }

  if (lane == 0) {
    float v1 = -INFINITY, v2 = -INFINITY;
    int   i1 = 0, i2 = 0;
#pragma unroll
    for (int e = 0; e < EE; ++e) {
      logits[(size_t)t * EE + e] = acc[e];
      if (acc[e] > v1)      { v2 = v1; i2 = i1; v1 = acc[e]; i1 = e; }
      else if (acc[e] > v2) { v2 = acc[e]; i2 = e; }
    }
    const float s1 = 1.f / (1.f + expf(-v1));
    const float s2 = 1.f / (1.f + expf(-v2));
#pragma unroll
    for (int e = 0; e < EE; ++e) scale[(size_t)t * EE + e] = 0.f;
    scale[(size_t)t * EE + i1] = s1;
    scale[(size_t)t * EE + i2] = s2;
    const int p1 = atomicAdd(&counts[i1], 1); lists[i1 * TT + p1] = t;
    const int p2 = atomicAdd(&counts[i2], 1); lists[i2 * TT + p2] = t;
  }
}

// ---------------------------------------------------------------------------
// Gated GEMM: Hout = silu(A @ Bg^T) * (A @ Bu^T)
// A:[M,Kd] (optionally row-gathered by idx and input-scaled), Bg/Bu:[N,Kd].
// Block tile 64x128, 8 waves (2x4), each wave owns a 32x32 register tile
// (8 accumulators), double-buffered LDS, one barrier per K-tile.
// ---------------------------------------------------------------------------
__global__ __launch_bounds__(256) void gemm_gated_kernel(
    const float* __restrict__ A, int lda, int Kd,
    const float* __restrict__ Bg, const float* __restrict__ Bu,
    float* __restrict__ Hout, int N,
    const int* __restrict__ cntp, const int* __restrict__ idx,
    const float* __restrict__ rscale /* = scale + e, stride EE; nullable */)
{
  __shared__ __align__(16) float Al [2][BM * LSTR];
  __shared__ __align__(16) float Bgl[2][BN * LSTR];
  __shared__ __align__(16) float Bul[2][BN * LSTR];

  const int cnt = cntp ? *cntp : TT;
  const int m0  = blockIdx.x * BM;
  if (m0 >= cnt) return;
  const int n0  = blockIdx.y * BN;
  const int tid = threadIdx.x;

  // Staging indices: each thread carries one float4 of A and two float4 of
  // each B matrix per K-tile.
  const int sr  = tid >> 2;        // 0..63
  const int ks4 = (tid & 3) << 2;  // 0,4,8,12

  long  arow = m0 + sr;
  float s    = 1.f;
  if (idx) {
    if (m0 + sr < cnt) {
      const int tkn = idx[m0 + sr];
      arow = tkn;
      if (rscale) s = rscale[(size_t)tkn * EE];
    } else arow = -1;
  }
  const float* bg0p = Bg + (size_t)(n0 + sr)      * Kd + ks4;
  const float* bg1p = Bg + (size_t)(n0 + sr + 64) * Kd + ks4;
  const float* bu0p = Bu + (size_t)(n0 + sr)      * Kd + ks4;
  const float* bu1p = Bu + (size_t)(n0 + sr + 64) * Kd + ks4;

  const int lane = tid & 31;
  const int wv   = tid >> 5;
  const int wm   = (wv & 1) << 5;   // 0 / 32
  const int wn   = (wv >> 1) << 5;  // 0 / 32 / 64 / 96
  const int mf   = lane & 15;
  const int ko   = (lane >> 4) << 1;   // K sub-offset 0 or 2 (ISA f32 A/B layout)

  v8f g00 = {}, g01 = {}, g10 = {}, g11 = {};
  v8f u00 = {}, u01 = {}, u10 = {}, u11 = {};

  // ---- prologue: stage K-tile 0 into LDS buffer 0
  float4 avr = make_float4(0.f, 0.f, 0.f, 0.f);
  if (arow >= 0) avr = *(const float4*)(A + (size_t)arow * lda + ks4);
  avr.x *= s; avr.y *= s; avr.z *= s; avr.w *= s;
  float4 g0r = *(const float4*)bg0p;
  float4 g1r = *(const float4*)bg1p;
  float4 u0r = *(const float4*)bu0p;
  float4 u1r = *(const float4*)bu1p;
  *(float4*)&Al [0][sr * LSTR + ks4]        = avr;
  *(float4*)&Bgl[0][sr * LSTR + ks4]        = g0r;
  *(float4*)&Bgl[0][(sr + 64) * LSTR + ks4] = g1r;
  *(float4*)&Bul[0][sr * LSTR + ks4]        = u0r;
  *(float4*)&Bul[0][(sr + 64) * LSTR + ks4] = u1r;
  __syncthreads();

  int cur = 0;
  for (int kb = 0; kb < Kd; kb += BK) {
    const bool has_next = (kb + BK) < Kd;
    if (has_next) {                       // global loads in flight during WMMAs
      avr = make_float4(0.f, 0.f, 0.f, 0.f);
      if (arow >= 0) avr = *(const float4*)(A + (size_t)arow * lda + kb + BK + ks4);
      avr.x *= s; avr.y *= s; avr.z *= s; avr.w *= s;
      g0r = *(const float4*)(bg0p + kb + BK);
      g1r = *(const float4*)(bg1p + kb + BK);
      u0r = *(const float4*)(bu0p + kb + BK);
      u1r = *(const float4*)(bu1p + kb + BK);
    }

    const float* Ac = Al [cur];
    const float* Gc = Bgl[cur];
    const float* Uc = Bul[cur];
#pragma unroll
    for (int kk = 0; kk < BK; kk += 4) {
      const v2f a0 = mk2(*(const float2*)&Ac[(wm      + mf) * LSTR + kk + ko]);
      const v2f a1 = mk2(*(const float2*)&Ac[(wm + 16 + mf) * LSTR + kk + ko]);
      const v2f bg0 = mk2(*(const float2*)&Gc[(wn      + mf) * LSTR + kk + ko]);
      const v2f bg1 = mk2(*(const float2*)&Gc[(wn + 16 + mf) * LSTR + kk + ko]);
      const v2f bu0 = mk2(*(const float2*)&Uc[(wn      + mf) * LSTR + kk + ko]);
      const v2f bu1 = mk2(*(const float2*)&Uc[(wn + 16 + mf) * LSTR + kk + ko]);
      g00 = WMMA_F32(a0, bg0, g00);  g01 = WMMA_F32(a0, bg1, g01);
      g10 = WMMA_F32(a1, bg0, g10);  g11 = WMMA_F32(a1, bg1, g11);
      u00 = WMMA_F32(a0, bu0, u00);  u01 = WMMA_F32(a0, bu1, u01);
      u10 = WMMA_F32(a1, bu0, u10);  u11 = WMMA_F32(a1, bu1, u11);
    }

    if (has_next) {
      const int nxt = cur ^ 1;
      *(float4*)&Al [nxt][sr * LSTR + ks4]        = avr;
      *(float4*)&Bgl[nxt][sr * LSTR + ks4]        = g0r;
      *(float4*)&Bgl[nxt][(sr + 64) * LSTR + ks4] = g1r;
      *(float4*)&Bul[nxt][sr * LSTR + ks4]        = u0r;
      *(float4*)&Bul[nxt][(sr + 64) * LSTR + ks4] = u1r;
    }
    __syncthreads();
    cur ^= 1;
  }

  // ---- epilogue: silu(g)*u, C layout: VGPR r -> M = r (lanes 0-15) / 8+r
#pragma unroll
  for (int r = 0; r < 8; ++r) {
    const int rm  = ((lane >> 4) << 3) + r;
    const int gm0 = m0 + wm + rm;
    const int gm1 = gm0 + 16;
    const int gn0 = n0 + wn + mf;
    const int gn1 = gn0 + 16;
    if (!idx || gm0 < cnt) {
      float g = g00[r], u = u00[r];
      Hout[(size_t)gm0 * N + gn0] = (g / (1.f + expf(-g))) * u;
      g = g01[r]; u = u01[r];
      Hout[(size_t)gm0 * N + gn1] = (g / (1.f + expf(-g))) * u;
    }
    if (!idx || gm1 < cnt) {
      float g = g10[r], u = u10[r];
      Hout[(size_t)gm1 * N + gn0] = (g / (1.f + expf(-g))) * u;
      g = g11[r]; u = u11[r];
      Hout[(size_t)gm1 * N + gn1] = (g / (1.f + expf(-g))) * u;
    }
  }
}

// ---------------------------------------------------------------------------
// Down GEMM: C = A @ Bd^T.  If idx: scatter-accumulate C rows into
// Out[idx[row]] (routed experts).  Else overwrite Out (shared expert; this
// also re-initializes d_out on every replay).  Same 64x128 / 32x32-per-wave
// double-buffered scheme, 4 accumulators.
// ---------------------------------------------------------------------------
__global__ __launch_bounds__(256) void gemm_down_kernel(
    const float* __restrict__ A, int lda, int Kd,
    const float* __restrict__ Bd,
    float* __restrict__ Out, int N,
    const int* __restrict__ cntp, const int* __restrict__ idx)
{
  __shared__ __align__(16) float Al[2][BM * LSTR];
  __shared__ __align__(16) float Bl[2][BN * LSTR];

  const int cnt = cntp ? *cntp : TT;
  const int m0  = blockIdx.x * BM;
  if (m0 >= cnt) return;
  const int n0  = blockIdx.y * BN;
  const int tid = threadIdx.x;

  const int sr  = tid >> 2;
  const int ks4 = (tid & 3) << 2;
  const long arow = (idx && (m0 + sr >= cnt)) ? -1 : (long)(m0 + sr); // A compacted
  const float* b0p = Bd + (size_t)(n0 + sr)      * Kd + ks4;
  const float* b1p = Bd + (size_t)(n0 + sr + 64) * Kd + ks4;

  const int lane = tid & 31;
  const int wv   = tid >> 5;
  const int wm   = (wv & 1) << 5;
  const int wn   = (wv >> 1) << 5;
  const int mf   = lane & 15;
  const int ko   = (lane >> 4) << 1;

  v8f c00 = {}, c01 = {}, c10 = {}, c11 = {};

  float4 avr = make_float4(0.f, 0.f, 0.f, 0.f);
  if (arow >= 0) avr = *(const float4*)(A + (size_t)arow * lda + ks4);
  float4 b0r = *(const float4*)b0p;
  float4 b1r = *(const float4*)b1p;
  *(float4*)&Al[0][sr * LSTR + ks4]        = avr;
  *(float4*)&Bl[0][sr * LSTR + ks4]        = b0r;
  *(float4*)&Bl[0][(sr + 64) * LSTR + ks4] = b1r;
  __syncthreads();

  int cur = 0;
  for (int kb = 0; kb < Kd; kb += BK) {
    const bool has_next = (kb + BK) < Kd;
    if (has_next) {
      avr = make_float4(0.f, 0.f, 0.f, 0.f);
      if (arow >= 0) avr = *(const float4*)(A + (size_t)arow * lda + kb + BK + ks4);
      b0r = *(const float4*)(b0p + kb + BK);
      b1r = *(const float4*)(b1p + kb + BK);
    }

    const float* Ac = Al[cur];
    const float* Bc = Bl[cur];
#pragma unroll
    for (int kk = 0; kk < BK; kk += 4) {
      const v2f a0 = mk2(*(const float2*)&Ac[(wm      + mf) * LSTR + kk + ko]);
      const v2f a1 = mk2(*(const float2*)&Ac[(wm + 16 + mf) * LSTR + kk + ko]);
      const v2f b0 = mk2(*(const float2*)&Bc[(wn      + mf) * LSTR + kk + ko]);
      const v2f b1 = mk2(*(const float2*)&Bc[(wn + 16 + mf) * LSTR + kk + ko]);
      c00 = WMMA_F32(a0, b0, c00);  c01 = WMMA_F32(a0, b1, c01);
      c10 = WMMA_F32(a1, b0, c10);  c11 = WMMA_F32(a1, b1, c11);
    }

    if (has_next) {
      const int nxt = cur ^ 1;
      *(float4*)&Al[nxt][sr * LSTR + ks4]        = avr;
      *(float4*)&Bl[nxt][sr * LSTR + ks4]        = b0r;
      *(float4*)&Bl[nxt][(sr + 64) * LSTR + ks4] = b1r;
    }
    __syncthreads();
    cur ^= 1;
  }

#pragma unroll
  for (int r = 0; r < 8; ++r) {
    const int rm  = ((lane >> 4) << 3) + r;
    const int gm0 = m0 + wm + rm;
    const int gm1 = gm0 + 16;
    const int gn0 = n0 + wn + mf;
    const int gn1 = gn0 + 16;
    if (idx) {
      if (gm0 < cnt) {
        const size_t row = (size_t)idx[gm0] * N;
        Out[row + gn0] += c00[r];          // unique element per kernel; stream-ordered
        Out[row + gn1] += c01[r];
      }
      if (gm1 < cnt) {
        const size_t row = (size_t)idx[gm1] * N;
        Out[row + gn0] += c10[r];
        Out[row + gn1] += c11[r];
      }
    } else {
      Out[(size_t)gm0 * N + gn0] = c00[r];
      Out[(size_t)gm0 * N + gn1] = c01[r];
      Out[(size_t)gm1 * N + gn0] = c10[r];
      Out[(size_t)gm1 * N + gn1] = c11[r];
    }
  }
}

// ---------------------------------------------------------------------------
extern "C" void kernel_launch(void* const* d_in, const int* in_sizes, int n_in,
                              void* d_out, int out_size, void* d_ws, size_t ws_size,
                              hipStream_t stream) {
  (void)in_sizes; (void)n_in; (void)out_size; (void)ws_size;

  const float* x   = (const float*)d_in[0];   // [B,S,H]
  const float* rw  = (const float*)d_in[1];   // [E,H]
  const float* gw  = (const float*)d_in[2];   // [E,I,H]
  const float* uw  = (const float*)d_in[3];   // [E,I,H]
  const float* dw  = (const float*)d_in[4];   // [E,H,I]
  const float* sgw = (const float*)d_in[5];   // [IS,H]
  const float* suw = (const float*)d_in[6];   // [IS,H]
  const float* sdw = (const float*)d_in[7];   // [H,IS]

  float* out    = (float*)d_out;              // [B,S,H]
  float* logits = out + (size_t)TT * HH;      // [B,S,E] (second tuple output)

  // workspace carve-up (256B aligned regions)
  char* p = (char*)d_ws;
  float* scale  = (float*)p;  p += ((size_t)TT * EE * 4 + 255) & ~255ULL;   // [T,E]
  int*   counts = (int*)p;    p += 256;                                      // [E]
  int*   lists  = (int*)p;    p += ((size_t)EE * TT * 4 + 255) & ~255ULL;   // [E,T]
  float* hs     = (float*)p;  p += ((size_t)TT * ISS * 4 + 255) & ~255ULL;  // [T,IS]
  float* hmid   = (float*)p;                                                 // [T,I] reused per expert

  hipMemsetAsync(counts, 0, EE * sizeof(int), stream);

  router_kernel<<<TT / 8, 256, 0, stream>>>(x, rw, logits, scale, counts, lists);

  // Shared expert: hs = silu(x@Wg^T)*(x@Wu^T); out = hs @ Wd^T (overwrite)
  gemm_gated_kernel<<<dim3(TT / BM, ISS / BN), 256, 0, stream>>>(
      x, HH, HH, sgw, suw, hs, ISS, nullptr, nullptr, nullptr);
  gemm_down_kernel<<<dim3(TT / BM, HH / BN), 256, 0, stream>>>(
      hs, ISS, ISS, sdw, out, HH, nullptr, nullptr);

  // Routed experts: gathered tokens, input scaled by sigmoid(router score)
  for (int e = 0; e < EE; ++e) {
    gemm_gated_kernel<<<dim3(TT / BM, II / BN), 256, 0, stream>>>(
        x, HH, HH,
        gw + (size_t)e * II * HH, uw + (size_t)e * II * HH,
        hmid, II, counts + e, lists + e * TT, scale + e);
    gemm_down_kernel<<<dim3(TT / BM, HH / BN), 256, 0, stream>>>(
        hmid, II, II, dw + (size_t)e * HH * II, out, HH,
        counts + e, lists + e * TT);
  }
}